// STFT_53068615909608
// MI455X (gfx1250) — compile-verified
//
#include <hip/hip_runtime.h>

// STFT-as-GEMM for MI455X (gfx1250), fp32 WMMA 16x16x4.
// O_re/im[b, m(bin), t(frame)] = sum_k W_re/im[m,k] * xp[b, t*HOP + k]
// xp is x reflect-padded by NFFT/2 on both sides (computed on the fly).

#define NFFT    2048
#define HOP     512
#define NBINS   1025
#define TFRAMES 1025
#define LSIG    524288
#define PADH    1024
#define NBATCH  32

#define BM 64        // bins per block
#define BN 64        // frames per block
#define KC 64        // K chunk per LDS stage
#define AROW 34      // float2 per LDS row: KC/2 = 32 + 2 pad (even stride -> conflict-free)

typedef float v2f __attribute__((ext_vector_type(2)));
typedef float v4f __attribute__((ext_vector_type(4)));
typedef float v8f __attribute__((ext_vector_type(8)));

__device__ __forceinline__ v8f wmma4(v2f a, v2f b, v8f c) {
#if defined(__HIP_DEVICE_COMPILE__)
  // (neg_a, A, neg_b, B, c_mod, C, reuse_a, reuse_b)
  return __builtin_amdgcn_wmma_f32_16x16x4_f32(false, a, false, b, (short)0, c,
                                               false, false);
#else
  // Host semantic-analysis stub; never executed.
  (void)a; (void)b;
  return c;
#endif
}

__global__ __launch_bounds__(256) void stft_wmma_kernel(
    const float* __restrict__ x,     // (32, LSIG)
    const float* __restrict__ wre,   // (NBINS, NFFT)
    const float* __restrict__ wim,   // (NBINS, NFFT)
    float* __restrict__ out)         // real (32,NBINS,T) then imag (32,NBINS,T)
{
  // k-pair layout: element [.,.,j] = {val(k=2j), val(k=2j+1)} -> one b64 load
  // is exactly a WMMA f32 A/B fragment (v0=even-k, v1=odd-k), with
  // j = 2*ks + (lane>=16).
  __shared__ v2f Asm[2][BM][AROW];   // [plane re/im][bin row][k pair]  34,816 B
  __shared__ v2f Bsm[BN][AROW];      // [frame][k pair]                 17,408 B

  const int tid  = threadIdx.x;
  const int lane = tid & 31;
  const int wid  = tid >> 5;       // 8 waves
  const int mi   = lane & 15;
  const int hi   = lane >> 4;      // half-wave select (K +2 for A/B, M +8 for C/D)

  const int mw = wid & 3;          // wave's 16-bin subtile  (4 along M)
  const int nw = wid >> 2;         // wave's 32-frame half   (2 along N)
  const int m0 = mw * 16;
  const int n0 = nw * 32;

  const int bx = blockIdx.x;       // bin tile   [0,17)
  const int ty = blockIdx.y;       // frame tile [0,17)
  const int bz = blockIdx.z;       // batch      [0,32)

  const float* xb = x + (size_t)bz * LSIG;

  v8f accRe[2], accIm[2];
  #pragma unroll
  for (int i = 0; i < 2; ++i) {
    accRe[i] = (v8f){0.f,0.f,0.f,0.f,0.f,0.f,0.f,0.f};
    accIm[i] = (v8f){0.f,0.f,0.f,0.f,0.f,0.f,0.f,0.f};
  }

  for (int kc = 0; kc < NFFT; kc += KC) {
    __syncthreads();   // previous chunk's fragment reads done before restage

    // ---- stage A: 2 planes x 64 bins x 64 k = 2048 float4, 8 per thread ----
    #pragma unroll
    for (int it = 0; it < 8; ++it) {
      int f4 = it * 256 + tid;         // [0,2048)
      int c  = f4 & 15;                // float4 within 64-k row
      int m  = (f4 >> 4) & 63;         // bin row in tile
      int p  = f4 >> 10;               // 0=real weights, 1=imag weights
      int mg = bx * BM + m;
      if (mg > NBINS - 1) mg = NBINS - 1;          // clamp, masked on store
      const float* wsrc = p ? wim : wre;
      v4f v = *(const v4f*)(wsrc + (size_t)mg * NFFT + kc + 4 * c);
      *(v4f*)((float*)&Asm[p][m][0] + 4 * c) = v;  // word idx 68*m + k
    }

    // ---- stage B: 64 frames x 64 k = 1024 float4, 4 per thread ----
    #pragma unroll
    for (int it = 0; it < 4; ++it) {
      int f4 = it * 256 + tid;         // [0,1024)
      int c  = f4 & 15;
      int n  = f4 >> 4;                // frame in tile
      long t  = (long)ty * BN + n;
      long j0 = t * HOP + kc + 4 * c - PADH;       // signal index (pre-reflect)
      v4f v;
      if (j0 >= 0 && j0 + 3 < LSIG) {              // fast path: aligned b128
        v = *(const v4f*)(xb + j0);
      } else {                                     // reflect/clamp boundaries
        #pragma unroll
        for (int e = 0; e < 4; ++e) {
          long j = j0 + e;
          if (j < 0)     j = -j;
          if (j >= LSIG) j = 2L * LSIG - 2 - j;
          if (j < 0)     j = 0;
          if (j >= LSIG) j = LSIG - 1;
          v[e] = xb[j];
        }
      }
      *(v4f*)((float*)&Bsm[n][0] + 4 * c) = v;
    }
    __syncthreads();

    // ---- compute: 16 K-steps, 4 WMMA + 4 ds_load_b64 per step ----
    const v2f* aRe = &Asm[0][m0 + mi][hi];
    const v2f* aIm = &Asm[1][m0 + mi][hi];
    const v2f* b0p = &Bsm[n0 + mi][hi];
    const v2f* b1p = &Bsm[n0 + 16 + mi][hi];
    #pragma unroll
    for (int ks = 0; ks < KC / 4; ++ks) {
      v2f are = aRe[2 * ks];
      v2f aim = aIm[2 * ks];
      v2f b0  = b0p[2 * ks];
      v2f b1  = b1p[2 * ks];
      accRe[0] = wmma4(are, b0, accRe[0]);
      accIm[0] = wmma4(aim, b0, accIm[0]);
      accRe[1] = wmma4(are, b1, accRe[1]);
      accIm[1] = wmma4(aim, b1, accIm[1]);
    }
  }

  // ---- store: C/D layout VGPR r -> M = r + 8*hi, N = lane&15 ----
  float* outRe = out;
  float* outIm = out + (size_t)NBATCH * NBINS * TFRAMES;
  #pragma unroll
  for (int nt = 0; nt < 2; ++nt) {
    int t = ty * BN + n0 + nt * 16 + mi;
    if (t < TFRAMES) {
      #pragma unroll
      for (int r = 0; r < 8; ++r) {
        int m = bx * BM + m0 + r + hi * 8;
        if (m < NBINS) {
          size_t o = ((size_t)bz * NBINS + m) * (size_t)TFRAMES + t;
          outRe[o] = accRe[nt][r];
          outIm[o] = accIm[nt][r];
        }
      }
    }
  }
}

extern "C" void kernel_launch(void* const* d_in, const int* in_sizes, int n_in,
                              void* d_out, int out_size, void* d_ws, size_t ws_size,
                              hipStream_t stream) {
  (void)in_sizes; (void)n_in; (void)out_size; (void)d_ws; (void)ws_size;
  const float* x   = (const float*)d_in[0];
  const float* wre = (const float*)d_in[1];
  const float* wim = (const float*)d_in[2];
  float* out = (float*)d_out;

  dim3 grid((NBINS + BM - 1) / BM,      // 17 bin tiles
            (TFRAMES + BN - 1) / BN,    // 17 frame tiles
            NBATCH);                    // 32 batches
  stft_wmma_kernel<<<grid, 256, 0, stream>>>(x, wre, wim, out);
}